// QuantumClassifier_91001767067764
// MI455X (gfx1250) — compile-verified
//
#include <hip/hip_runtime.h>
#include <math.h>

typedef __attribute__((ext_vector_type(2))) float v2f;
typedef __attribute__((ext_vector_type(8))) float v8f;

#define CONV_THREADS 128            // 4 waves (wave32), 64 samples / block
#define SAMP_PER_BLOCK 64
#define KREAL 189                   // 7*27 flattened image = GEMM K dim
#define KPAD  192                   // padded to multiple of 4 for 16x16x4 WMMA

// ---------------------------------------------------------------------------
// Kernel 1: conv as GEMM  out(B x 5) = X(B x 189) * Weff(189 x 5)  via
// V_WMMA_F32_16X16X4_F32.  Each wave owns a 16-sample tile, 48 K-steps,
// dual accumulator chains for matrix-pipe ILP.
// ---------------------------------------------------------------------------
__global__ __launch_bounds__(CONV_THREADS)
void conv_wmma_kernel(const float* __restrict__ x,
                      const float* __restrict__ w,
                      const float* __restrict__ bias,
                      float* __restrict__ y)
{
    __shared__ __align__(16) float s_x[SAMP_PER_BLOCK * KREAL + 8]; // 48.4 KB
    __shared__ float s_w[KPAD * 16];                                 // 12 KB

    const int tid = threadIdx.x;

    // Build Weff[k][n]: w[r, cc-5n] for k=r*27+cc, zero outside window / pads.
    for (int idx = tid; idx < KPAD * 16; idx += CONV_THREADS) {
        int k = idx >> 4, n = idx & 15;
        float val = 0.0f;
        if (n < 5 && k < KREAL) {
            int r = k / 27, cc = k % 27;
            int cl = cc - 5 * n;
            if (cl >= 0 && cl < 7) val = w[r * 7 + cl];
        }
        s_w[idx] = val;
    }

    // Coalesced float4 staging of 64 samples (64*189 floats, 16B-aligned).
    const float4* xin = (const float4*)(x + (size_t)blockIdx.x * (SAMP_PER_BLOCK * KREAL));
    float4* sx4 = (float4*)s_x;
    for (int i = tid; i < (SAMP_PER_BLOCK * KREAL) / 4; i += CONV_THREADS)
        sx4[i] = xin[i];

    __syncthreads();

    const int lane = tid & 31;
    const int wave = tid >> 5;
    const int m    = lane & 15;     // A: row M.  B: col N.
    const int hi   = lane >> 4;     // half-wave -> K offset (+2) / M offset (+8 in D)
    const int srow = wave * 16;
    const int abase = (srow + m) * KREAL + 2 * hi;
    const int bbase = (2 * hi) * 16 + m;

    v8f acc0 = {};
    v8f acc1 = {};
    #pragma unroll
    for (int step = 0; step < 48; ++step) {
        const int ka = step * 4;
        v2f av, bv;
        av.x = s_x[abase + ka];                    // A(m, k)   (k>=189 reads are
        av.y = s_x[abase + ka + 1];                //  killed by zero Weff rows)
        bv.x = s_w[bbase + ka * 16];               // B(k, n=m)
        bv.y = s_w[bbase + (ka + 1) * 16];
        if (step & 1)
            acc1 = __builtin_amdgcn_wmma_f32_16x16x4_f32(
                false, av, false, bv, (short)0, acc1, false, false);
        else
            acc0 = __builtin_amdgcn_wmma_f32_16x16x4_f32(
                false, av, false, bv, (short)0, acc0, false, false);
    }

    if (m < 5) {                                   // only N=0..4 are real outputs
        const float b0 = bias[0];
        const size_t gbase = (size_t)blockIdx.x * SAMP_PER_BLOCK + srow + 8 * hi;
        #pragma unroll
        for (int r = 0; r < 8; ++r)                // D VGPR r -> row M=r+8*hi
            y[(gbase + r) * 5 + m] = acc0[r] + acc1[r] + b0;
    }
}

// ---------------------------------------------------------------------------
// Kernels 2/3: deterministic two-stage reduction for batch mean / var.
// ---------------------------------------------------------------------------
__global__ __launch_bounds__(256)
void reduce_stage1(const float* __restrict__ y, float* __restrict__ part, int ntot)
{
    __shared__ float s0[256], s1[256];
    const int tid = threadIdx.x;
    const int base = blockIdx.x * 1024;
    float s = 0.f, q = 0.f;
    for (int i = tid; i < 1024; i += 256) {
        int g = base + i;
        if (g < ntot) { float v = y[g]; s += v; q += v * v; }
    }
    s0[tid] = s; s1[tid] = q; __syncthreads();
    for (int off = 128; off > 0; off >>= 1) {
        if (tid < off) { s0[tid] += s0[tid + off]; s1[tid] += s1[tid + off]; }
        __syncthreads();
    }
    if (tid == 0) { part[2 * blockIdx.x] = s0[0]; part[2 * blockIdx.x + 1] = s1[0]; }
}

__global__ __launch_bounds__(256)
void reduce_stage2(const float* __restrict__ part, float* __restrict__ stats,
                   int nb, int ntot)
{
    __shared__ float s0[256], s1[256];
    const int tid = threadIdx.x;
    float s = 0.f, q = 0.f;
    for (int i = tid; i < nb; i += 256) { s += part[2 * i]; q += part[2 * i + 1]; }
    s0[tid] = s; s1[tid] = q; __syncthreads();
    for (int off = 128; off > 0; off >>= 1) {
        if (tid < off) { s0[tid] += s0[tid + off]; s1[tid] += s1[tid + off]; }
        __syncthreads();
    }
    if (tid == 0) {
        float mean = s0[0] / (float)ntot;
        stats[0] = mean;
        stats[1] = s1[0] / (float)ntot - mean * mean;   // biased var (jnp default)
    }
}

// ---------------------------------------------------------------------------
// Kernel 4: 5-qubit state-vector circuit, one thread per sample.
// wire w acts on index bit (4-w).  All gate loops fully unrolled with
// compile-time masks so the 32 complex amplitudes live in VGPRs.
// ---------------------------------------------------------------------------
template<int MASK>
__device__ __forceinline__ void rx_gate(float re[32], float im[32], float c, float s)
{
    // RX: new0 = c*a0 - i*s*a1 ; new1 = -i*s*a0 + c*a1   (s = sin(theta/2))
    #pragma unroll
    for (int i = 0; i < 32; ++i) {
        if ((i & MASK) == 0) {
            const int j = i | MASK;
            float a0r = re[i], a0i = im[i], a1r = re[j], a1i = im[j];
            re[i] = c * a0r + s * a1i;
            im[i] = c * a0i - s * a1r;
            re[j] = c * a1r + s * a0i;
            im[j] = c * a1i - s * a0r;
        }
    }
}

template<int CMASK, int TMASK>
__device__ __forceinline__ void cnot_gate(float re[32], float im[32])
{
    #pragma unroll
    for (int i = 0; i < 32; ++i) {
        if ((i & CMASK) != 0 && (i & TMASK) == 0) {
            const int j = i | TMASK;
            float t;
            t = re[i]; re[i] = re[j]; re[j] = t;
            t = im[i]; im[i] = im[j]; im[j] = t;
        }
    }
}

template<int MA, int MB>
__device__ __forceinline__ void entangler(float re[32], float im[32],
                                          const float* wc, const float* ws)
{
    #pragma unroll
    for (int l = 0; l < 4; ++l) {
        rx_gate<MA>(re, im, wc[2 * l], ws[2 * l]);
        rx_gate<MB>(re, im, wc[2 * l + 1], ws[2 * l + 1]);
        cnot_gate<MA, MB>(re, im);
    }
}

__global__ __launch_bounds__(256)
void circuit_kernel(const float* __restrict__ y,
                    const float* __restrict__ stats,
                    const float* __restrict__ gamma,
                    const float* __restrict__ beta,
                    const float* __restrict__ w1,
                    const float* __restrict__ w2,
                    const float* __restrict__ w3,
                    float* __restrict__ out, int B)
{
    __shared__ float s_wc[24], s_ws[24];     // cos/sin of fixed entangler angles
    const int tid = threadIdx.x;
    if (tid < 24) {
        const float* wp = (tid < 8) ? w1 : ((tid < 16) ? w2 : w3);
        float sn, cs;
        sincosf(wp[tid & 7] * 0.5f, &sn, &cs);
        s_wc[tid] = cs; s_ws[tid] = sn;
    }
    __syncthreads();

    const int b = blockIdx.x * 256 + tid;
    if (b >= B) return;

    const float mean   = stats[0];
    const float invstd = rsqrtf(stats[1] + 1e-5f);
    const float g = gamma[0], bt = beta[0];

    float fc[5], fs[5];
    #pragma unroll
    for (int q = 0; q < 5; ++q) {
        float f = (y[b * 5 + q] - mean) * invstd * g + bt;   // BN'd feature
        sincosf(0.5f * f, &fs[q], &fc[q]);
    }

    float re[32], im[32];
    #pragma unroll
    for (int i = 0; i < 32; ++i) { re[i] = 0.f; im[i] = 0.f; }
    re[0] = 1.f;

    // Initial data-dependent RX on wires 0..4 (bit masks 16,8,4,2,1)
    rx_gate<16>(re, im, fc[0], fs[0]);
    rx_gate< 8>(re, im, fc[1], fs[1]);
    rx_gate< 4>(re, im, fc[2], fs[2]);
    rx_gate< 2>(re, im, fc[3], fs[3]);
    rx_gate< 1>(re, im, fc[4], fs[4]);

    entangler<16, 8>(re, im, s_wc +  0, s_ws +  0);   // w1 on wires (0,1)
    entangler< 4, 2>(re, im, s_wc +  8, s_ws +  8);   // w2 on wires (2,3)
    entangler< 8, 4>(re, im, s_wc + 16, s_ws + 16);   // w3 on wires (1,2)

    cnot_gate<16, 1>(re, im);                          // CNOT(c,4), c=0..3
    cnot_gate< 8, 1>(re, im);
    cnot_gate< 4, 1>(re, im);
    cnot_gate< 2, 1>(re, im);

    float p0 = 0.f, p1 = 0.f;
    #pragma unroll
    for (int i = 0; i < 32; i += 2) {
        p0 += re[i] * re[i] + im[i] * im[i];
        p1 += re[i + 1] * re[i + 1] + im[i + 1] * im[i + 1];
    }
    const float mx = fmaxf(p0, p1);
    const float l  = mx + logf(expf(p0 - mx) + expf(p1 - mx));
    ((float2*)out)[b] = make_float2(p0 - l, p1 - l);
}

// ---------------------------------------------------------------------------
extern "C" void kernel_launch(void* const* d_in, const int* in_sizes, int n_in,
                              void* d_out, int out_size, void* d_ws, size_t ws_size,
                              hipStream_t stream)
{
    const float* x      = (const float*)d_in[0];
    const float* conv_w = (const float*)d_in[1];
    const float* conv_b = (const float*)d_in[2];
    const float* gamma  = (const float*)d_in[3];
    const float* beta   = (const float*)d_in[4];
    const float* w1     = (const float*)d_in[5];
    const float* w2     = (const float*)d_in[6];
    const float* w3     = (const float*)d_in[7];
    float* out = (float*)d_out;

    const int B    = in_sizes[0] / (7 * 27);   // 131072
    const int ntot = B * 5;
    const int nb1  = (ntot + 1023) / 1024;     // 640 stage-1 blocks

    float* y_ws  = (float*)d_ws;               // B*5 conv outputs (2.6 MB)
    float* part  = y_ws + (size_t)ntot;        // 2*nb1 partials
    float* stats = part + 2 * (size_t)nb1;     // {mean, var}

    conv_wmma_kernel<<<B / SAMP_PER_BLOCK, CONV_THREADS, 0, stream>>>(x, conv_w, conv_b, y_ws);
    reduce_stage1<<<nb1, 256, 0, stream>>>(y_ws, part, ntot);
    reduce_stage2<<<1, 256, 0, stream>>>(part, stats, nb1, ntot);
    circuit_kernel<<<(B + 255) / 256, 256, 0, stream>>>(y_ws, stats, gamma, beta, w1, w2, w3, out, B);
}